// DynamicConv_5970004542047
// MI455X (gfx1250) — compile-verified
//
#include <hip/hip_runtime.h>

typedef float v2f __attribute__((ext_vector_type(2)));
typedef float v8f __attribute__((ext_vector_type(8)));

#define CCH   128   // channels
#define IMG   96    // H == W
#define HWPIX 9216  // 96*96
#define LSTR  130   // LDS row stride in dwords (even -> 8B-aligned b64 loads)

// Fused: dynamic depthwise 3x3 -> 1x1 pointwise (WMMA f32 GEMM) -> BN -> ReLU.
// One workgroup per (batch, image row): 128 ch x 96 pixels.
__global__ __launch_bounds__(256) void dynconv_fused(
    const float* __restrict__ x,   // [B,C,96,96]
    const float* __restrict__ kf,  // [B,C,3,3]
    const float* __restrict__ Wp,  // [C,C] pointwise weight (row = out ch)
    const float* __restrict__ g,   // bn_gamma [C]
    const float* __restrict__ bt,  // bn_beta  [C]
    const float* __restrict__ mn,  // bn_mean  [C]
    const float* __restrict__ vr,  // bn_var   [C]
    float* __restrict__ out)       // [B,C,96,96]
{
  __shared__ float Xs[IMG * LSTR];  // depthwise result: [pixel-in-row][channel]
  __shared__ float Ks[CCH * 9];     // per-sample depthwise kernels

  const int b    = blockIdx.y;
  const int row0 = blockIdx.x;
  const int t    = threadIdx.x;

  // --- stage 0: cache k[b] in LDS ---
  for (int i = t; i < CCH * 9; i += 256) Ks[i] = kf[b * CCH * 9 + i];
  __syncthreads();

  // --- stage 1: depthwise 3x3 for this row, all 128 channels, into LDS ---
  for (int i = t; i < CCH * IMG; i += 256) {
    const int c = i / IMG;
    const int w = i - c * IMG;
    const float* xp = x + (size_t)(b * CCH + c) * HWPIX;
    const float* kp = Ks + c * 9;
    float acc = 0.f;
#pragma unroll
    for (int di = -1; di <= 1; ++di) {
      const int hh = row0 + di;
      if ((unsigned)hh < (unsigned)IMG) {
#pragma unroll
        for (int dj = -1; dj <= 1; ++dj) {
          const int ww = w + dj;
          if ((unsigned)ww < (unsigned)IMG)
            acc = fmaf(xp[hh * IMG + ww], kp[(di + 1) * 3 + (dj + 1)], acc);
        }
      }
    }
    Xs[w * LSTR + c] = acc;  // bank stride 2 -> conflict-free across 32 lanes
  }
  __syncthreads();

  // --- stage 2: Y = W x Xnew via V_WMMA_F32_16X16X4_F32, fused BN+ReLU ---
  const int wv   = t >> 5;        // wave id 0..7 -> owns out channels 16*wv..+15
  const int lane = t & 31;
  const int nlo  = lane & 15;     // A row (M) / B col (N) for this lane
  const int cb   = (lane < 16) ? 0 : 2;  // K sub-offset per 16x4 fp32 layout

  // Preload all 32 K-step A fragments (weights) -- reused across all N-tiles.
  v2f aw[32];
  const float* Wr = Wp + (16 * wv + nlo) * CCH + cb;
#pragma unroll
  for (int kk = 0; kk < 32; ++kk)
    aw[kk] = *(const v2f*)(Wr + 4 * kk);

  // BN scale/bias for the channels this lane's 8 accumulator regs map to:
  // reg r -> channel chb + r (C/D layout: lanes 0-15 -> M=r, lanes 16-31 -> M=r+8)
  const int chb = 16 * wv + ((lane < 16) ? 0 : 8);
  float sc[8], bi[8];
#pragma unroll
  for (int r = 0; r < 8; ++r) {
    const int ch   = chb + r;
    const float iv = g[ch] * rsqrtf(vr[ch] + 1e-5f);
    sc[r] = iv;
    bi[r] = bt[ch] - mn[ch] * iv;
  }

  // 6 N-tiles of 16 pixels, processed 2 at a time: two independent
  // accumulator chains per wave -> 2x WMMA issue-level parallelism, and each
  // A fragment feeds two back-to-back WMMAs.
  for (int np = 0; np < 3; ++np) {
    v8f acc0 = {};
    v8f acc1 = {};
    const float* bp0 = &Xs[((2 * np    ) * 16 + nlo) * LSTR + cb];
    const float* bp1 = &Xs[((2 * np + 1) * 16 + nlo) * LSTR + cb];
#pragma unroll
    for (int kk = 0; kk < 32; ++kk) {    // K = 128 channels, 4 per wmma
      v2f b0 = *(const v2f*)(bp0 + 4 * kk);  // ds_load_b64
      v2f b1 = *(const v2f*)(bp1 + 4 * kk);
      acc0 = __builtin_amdgcn_wmma_f32_16x16x4_f32(
          false, aw[kk], false, b0, (short)0, acc0, false, false);
      acc1 = __builtin_amdgcn_wmma_f32_16x16x4_f32(
          false, aw[kk], false, b1, (short)0, acc1, false, false);
    }
    const int p0 = row0 * IMG + (2 * np) * 16 + nlo;
    float* op0 = out + ((size_t)b * CCH + chb) * HWPIX + p0;
    float* op1 = op0 + 16;  // next N-tile, same channels
#pragma unroll
    for (int r = 0; r < 8; ++r) {
      const float v0 = fmaf(acc0[r], sc[r], bi[r]);
      const float v1 = fmaf(acc1[r], sc[r], bi[r]);
      op0[r * HWPIX] = v0 > 0.f ? v0 : 0.f;  // BN + ReLU fused into the store
      op1[r * HWPIX] = v1 > 0.f ? v1 : 0.f;
    }
  }
}

extern "C" void kernel_launch(void* const* d_in, const int* in_sizes, int n_in,
                              void* d_out, int out_size, void* d_ws, size_t ws_size,
                              hipStream_t stream) {
  const float* x  = (const float*)d_in[0];  // x        [32,128,96,96]
  const float* kf = (const float*)d_in[1];  // k        [32,128,3,3]
  const float* Wp = (const float*)d_in[2];  // pw_weight[128,128]
  const float* g  = (const float*)d_in[3];  // bn_gamma [128]
  const float* bt = (const float*)d_in[4];  // bn_beta  [128]
  const float* mn = (const float*)d_in[5];  // bn_mean  [128]
  const float* vr = (const float*)d_in[6];  // bn_var   [128]
  float* out = (float*)d_out;

  dim3 grid(IMG, 32);  // (image row, batch)
  dynconv_fused<<<grid, 256, 0, stream>>>(x, kf, Wp, g, bt, mn, vr, out);

  (void)in_sizes; (void)n_in; (void)out_size; (void)d_ws; (void)ws_size;
}